// OurConvTranspose3d_34041910788790
// MI455X (gfx1250) — compile-verified
//
#include <hip/hip_runtime.h>

typedef float v2f __attribute__((ext_vector_type(2)));
typedef float v8f __attribute__((ext_vector_type(8)));

#define DD 66
#define HHW (66 * 66)        /* 4356  */
#define DHW (66 * 66 * 66)   /* 287496 */
#define CI 32
#define CO 32

__global__ void zero_f4_kernel(float4* __restrict__ p, int n4) {
  int i = blockIdx.x * blockDim.x + threadIdx.x;
  int stride = gridDim.x * blockDim.x;
  for (; i < n4; i += stride) p[i] = make_float4(0.f, 0.f, 0.f, 0.f);
}

__launch_bounds__(256)
__global__ void p2g_wmma_kernel(const float* __restrict__ x,
                                const float* __restrict__ pos,
                                const float* __restrict__ kern,
                                const float* __restrict__ bias,
                                float* __restrict__ out,
                                int Mtotal, int Nper) {
  __shared__ float sB[1024];      // one offset's 32x32 kernel slice, B-fragment layout
  __shared__ float2 sP[8][32];    // per-wave: (weight, flat index) per particle

  const int t    = threadIdx.x;
  const int lane = t & 31;
  const int wv   = t >> 5;
  const int half = lane >> 4;     // 0: K lo pair, 1: K hi pair
  const int nlo  = lane & 15;     // N / M-within-tile coordinate

  const int waveBase = (blockIdx.x * 8 + wv) * 32;

  // ---- per-lane particle setup (lane owns particle waveBase+lane) ----
  int p  = waveBase + lane;
  int pc = (p < Mtotal) ? p : (Mtotal - 1);
  float valid = (p < Mtotal) ? 1.0f : 0.0f;

  float px = pos[pc * 3 + 0] * 64.0f;
  float py = pos[pc * 3 + 1] * 64.0f;
  float pz = pos[pc * 3 + 2] * 64.0f;
  int b0 = (int)px, b1 = (int)py, b2 = (int)pz;
  float fx = px - (float)b0 - 0.5f;
  float fy = py - (float)b1 - 0.5f;
  float fz = pz - (float)b2 - 0.5f;

  float wx[3], wy[3], wz[3];
  wx[0] = 0.5f * (0.5f - fx) * (0.5f - fx) * valid;   // fold tail mask into dim-0
  wx[1] = (0.75f - fx * fx) * valid;
  wx[2] = 0.5f * (0.5f + fx) * (0.5f + fx) * valid;
  wy[0] = 0.5f * (0.5f - fy) * (0.5f - fy);
  wy[1] = 0.75f - fy * fy;
  wy[2] = 0.5f * (0.5f + fy) * (0.5f + fy);
  wz[0] = 0.5f * (0.5f - fz) * (0.5f - fz);
  wz[1] = 0.75f - fz * fz;
  wz[2] = 0.5f * (0.5f + fz) * (0.5f + fz);

  int bb = (pc >= Nper) ? 1 : 0;
  int ibase = bb * (CO * DHW) + b0 * HHW + b1 * 66 + b2;

  // ---- A fragments: 2 M-tiles x 8 K-chunks, v2f per lane ----
  // ISA 32-bit A 16x4 layout: M = lane%16; lanes<16 -> K {0,1}, lanes>=16 -> K {2,3}
  v2f a[2][8];
#pragma unroll
  for (int mt = 0; mt < 2; ++mt) {
    int row = waveBase + mt * 16 + nlo;
    row = (row < Mtotal) ? row : (Mtotal - 1);
    const float* xr = x + row * CI + half * 2;
#pragma unroll
    for (int c = 0; c < 8; ++c) a[mt][c] = *(const v2f*)(xr + c * 4);
  }

  const float bl0 = bias[nlo];        // N-tile 0 channel bias
  const float bl1 = bias[16 + nlo];   // N-tile 1 channel bias
  const int chanOff0 = nlo * DHW;
  const int chanOff1 = (16 + nlo) * DHW;

  const float4* k4 = (const float4*)kern;

  for (int o = 0; o < 27; ++o) {
    const int oi = o / 9, oj = (o / 3) % 3, ok = o % 3;

    __syncthreads();   // previous iteration's sB/sP readers done

    // ---- cooperative transpose of K[o] (32x32) into B-fragment layout ----
    // dest slot for K[o][ci][co]:
    //   nt=co>>4, c=ci>>2, h=(ci>>1)&1, v=ci&1, lane=h*16+(co&15)
    //   sB[(nt*8+c)*64 + lane*2 + v]
    {
      float4 v4 = k4[o * 256 + t];          // 4 consecutive co, same ci
      int f   = t * 4;
      int ci  = f >> 5;
      int co  = f & 31;
      int nt  = co >> 4;
      int cc  = ci >> 2;
      int hh  = (ci >> 1) & 1;
      int vv  = ci & 1;
      int base = (nt * 8 + cc) * 64 + (hh * 16 + (co & 15)) * 2 + vv;
      sB[base + 0] = v4.x;
      sB[base + 2] = v4.y;
      sB[base + 4] = v4.z;
      sB[base + 6] = v4.w;
    }
    // per-particle (weight, flat-index) for this offset
    {
      float wp = wx[oi] * wy[oj] * wz[ok];
      int io = ibase + oi * HHW + oj * 66 + ok;
      sP[wv][lane] = make_float2(wp, __int_as_float(io));
    }
    __syncthreads();

    // ---- B fragments from LDS (conflict-free b64: lane stride 8B) ----
    v2f bf[2][8];
#pragma unroll
    for (int nt = 0; nt < 2; ++nt)
#pragma unroll
      for (int c = 0; c < 8; ++c)
        bf[nt][c] = *(const v2f*)&sB[(nt * 8 + c) * 64 + lane * 2];

    // ---- accumulate 16x16 tiles: d[mt][nt] = A[mt] * K[o][:,nt] + bias ----
    v8f dacc[2][2];
#pragma unroll
    for (int mt = 0; mt < 2; ++mt)
#pragma unroll
      for (int nt = 0; nt < 2; ++nt) {
        float bl = nt ? bl1 : bl0;
#pragma unroll
        for (int r = 0; r < 8; ++r) dacc[mt][nt][r] = bl;
      }

#pragma unroll
    for (int c = 0; c < 8; ++c) {
#pragma unroll
      for (int mt = 0; mt < 2; ++mt) {
#pragma unroll
        for (int nt = 0; nt < 2; ++nt) {
          dacc[mt][nt] = __builtin_amdgcn_wmma_f32_16x16x4_f32(
              false, a[mt][c], false, bf[nt][c], (short)0, dacc[mt][nt],
              false, false);
        }
      }
    }

    // ---- weighted scatter: C/D layout VGPR r <-> M = r (+8 for hi lanes) ----
#pragma unroll
    for (int mt = 0; mt < 2; ++mt) {
#pragma unroll
      for (int r = 0; r < 8; ++r) {
        int m = mt * 16 + half * 8 + r;
        float2 pw = sP[wv][m];
        float w = pw.x;
        int io = __float_as_int(pw.y);
        atomicAdd(out + io + chanOff0, w * dacc[mt][0][r]);
        atomicAdd(out + io + chanOff1, w * dacc[mt][1][r]);
      }
    }
  }
}

extern "C" void kernel_launch(void* const* d_in, const int* in_sizes, int n_in,
                              void* d_out, int out_size, void* d_ws, size_t ws_size,
                              hipStream_t stream) {
  const float* x    = (const float*)d_in[0];
  const float* pos  = (const float*)d_in[1];
  const float* kern = (const float*)d_in[2];
  const float* bias = (const float*)d_in[3];
  float* out = (float*)d_out;

  const int Mtotal = in_sizes[0] / CI;     // B*N = 400000
  const int Nper   = in_sizes[1] / (2 * 3);  // N = 200000 (B=2 per reference)

  // zero the padded output grid (2*32*66^3 floats)
  int n4 = out_size / 4;
  zero_f4_kernel<<<2048, 256, 0, stream>>>((float4*)out, n4);

  int blocks = (Mtotal + 255) / 256;       // 256 particles per block (8 waves)
  p2g_wmma_kernel<<<blocks, 256, 0, stream>>>(x, pos, kern, bias, out,
                                              Mtotal, Nper);
}